// GaussianVideo_Layer_22582938042783
// MI455X (gfx1250) — compile-verified
//
#include <hip/hip_runtime.h>
#include <math.h>

typedef __attribute__((ext_vector_type(2))) float v2f;
typedef __attribute__((ext_vector_type(4))) float v4f;
typedef __attribute__((ext_vector_type(8))) float v8f;

#define NG   1024
#define Wd   128
#define Hd   128
#define Td   8

// ---------------------------------------------------------------------------
// Prep: per-gaussian quadratic-form coefficients, stored PRE-PAIRED for the
// WMMA B-fragment layout:  Gp[(chunk*2 + half)][n] = (k, k+1) float pair with
// k = 4*chunk + 2*half.  K padded 10 -> 12.  Features opacity-folded and
// padded to float4 for a single b128 load.
// ---------------------------------------------------------------------------
__global__ __launch_bounds__(256) void prep_kernel(
    const float* __restrict__ xyz,       // [N][3]
    const float* __restrict__ chol,      // [N][6]
    const float* __restrict__ features,  // [N][3]
    const float* __restrict__ opacity,   // [N][1]
    v2f*  __restrict__ Gp,               // [6][N] float2
    v4f*  __restrict__ feat4)            // [N] float4
{
    int n = blockIdx.x * blockDim.x + threadIdx.x;
    if (n >= NG) return;

    // means = 0.5*((tanh(xyz)+1)*S - 1), S = (W,H,T)
    float mx = 0.5f * ((tanhf(xyz[n*3+0]) + 1.0f) * (float)Wd - 1.0f);
    float my = 0.5f * ((tanhf(xyz[n*3+1]) + 1.0f) * (float)Hd - 1.0f);
    float mt = 0.5f * ((tanhf(xyz[n*3+2]) + 1.0f) * (float)Td - 1.0f);

    // cholesky + bound [0.5, 0, 0.5, 0.5, 0, 0.5]
    float c0 = chol[n*6+0] + 0.5f;  // L00
    float c1 = chol[n*6+1];         // L10
    float c2 = chol[n*6+2] + 0.5f;  // L20
    float c3 = chol[n*6+3] + 0.5f;  // L11
    float c4 = chol[n*6+4];         // L21
    float c5 = chol[n*6+5] + 0.5f;  // L22

    // Sigma = L L^T (symmetric 3x3)
    float s00 = c0*c0;
    float s01 = c0*c1;
    float s02 = c0*c2;
    float s11 = c1*c1 + c3*c3;
    float s12 = c1*c2 + c3*c4;
    float s22 = c2*c2 + c4*c4 + c5*c5;

    // Q = Sigma^-1 via adjugate / det
    float A00 = s11*s22 - s12*s12;
    float A01 = s02*s12 - s01*s22;
    float A02 = s01*s12 - s02*s11;
    float A11 = s00*s22 - s02*s02;
    float A12 = s01*s02 - s00*s12;
    float A22 = s00*s11 - s01*s01;
    float det = s00*A00 + s01*A01 + s02*A02;
    float id  = 1.0f / det;
    float q00 = A00*id, q01 = A01*id, q02 = A02*id;
    float q11 = A11*id, q12 = A12*id, q22 = A22*id;

    // qm = Q m ;  const = 0.5 m^T Q m
    float qm0 = q00*mx + q01*my + q02*mt;
    float qm1 = q01*mx + q11*my + q12*mt;
    float qm2 = q02*mx + q12*my + q22*mt;
    float cst = 0.5f * (mx*qm0 + my*qm1 + mt*qm2);

    // Quadratic-form coefficient vector (K order matches F features)
    float gv[12] = { 0.5f*q00, 0.5f*q11, 0.5f*q22,
                     q01, q02, q12,
                     -qm0, -qm1, -qm2, cst, 0.0f, 0.0f };

#pragma unroll
    for (int c = 0; c < 3; ++c)
#pragma unroll
        for (int h = 0; h < 2; ++h) {
            v2f pr;
            pr.x = gv[4*c + 2*h + 0];
            pr.y = gv[4*c + 2*h + 1];
            Gp[(c*2 + h)*NG + n] = pr;
        }

    float op = opacity[n];
    v4f fv;
    fv.x = features[n*3+0] * op;
    fv.y = features[n*3+1] * op;
    fv.z = features[n*3+2] * op;
    fv.w = 0.0f;
    feat4[n] = fv;
}

// ---------------------------------------------------------------------------
// Fused rasterizer: one wave32 owns a 16-pixel tile, loops over 64 tiles of
// 16 gaussians. sigma via 3x chained V_WMMA_F32_16X16X4_F32 (K=12), then
// exp + rgb accumulation in registers, LDS column-reduction, permuted store.
// Per gaussian tile: 3x global_load_b64 (B) + 1x global_load_b128 (features).
// ---------------------------------------------------------------------------
__global__ __launch_bounds__(256) void render_kernel(
    const v2f* __restrict__ Gp,     // [6][N] float2
    const v4f* __restrict__ feat4,  // [N] float4
    float* __restrict__ out)        // (1,3,H,W,T) flat
{
    __shared__ float lds[8][16*3*16];   // per-wave: [pixRow 16][ch 3][col 16]

    const int lane  = threadIdx.x & 31;
    const int wv    = threadIdx.x >> 5;
    const int wave  = blockIdx.x * 8 + wv;
    const int pbase = wave * 16;        // first pixel of this tile
    const int half  = lane >> 4;        // 0: K0/K1 + rows 0-7 ; 1: K2/K3 + rows 8-15
    const int col   = lane & 15;        // A row index (M) and B/D column (gaussian)

    // Per-lane F features for pixel p = pbase + col (t-major flat pixel order)
    const int p  = pbase + col;
    const float x = (float)(p & (Wd-1));
    const float y = (float)((p >> 7) & (Hd-1));
    const float t = (float)(p >> 14);
    const float f[12] = { x*x, y*y, t*t, x*y, x*t, y*t, x, y, t, 1.0f, 0.0f, 0.0f };

    // A-matrix fragments (16x4 f32): lane<16 holds K = chunk*4 + {0,1},
    // lane>=16 holds K = chunk*4 + {2,3}  (ISA §7.12.2 layout).
    v2f a0, a1, a2;
    a0.x = f[0 + 2*half];  a0.y = f[1 + 2*half];
    a1.x = f[4 + 2*half];  a1.y = f[5 + 2*half];
    a2.x = f[8 + 2*half];  a2.y = f[9 + 2*half];

    // Per-chunk B base pointers for this lane's half
    const v2f* __restrict__ B0 = Gp + (0 + half)*NG;
    const v2f* __restrict__ B1 = Gp + (2 + half)*NG;
    const v2f* __restrict__ B2 = Gp + (4 + half)*NG;

    float rgb[8][3];
#pragma unroll
    for (int r = 0; r < 8; ++r) { rgb[r][0] = 0.f; rgb[r][1] = 0.f; rgb[r][2] = 0.f; }

    for (int nt = 0; nt < NG/16; ++nt) {
        const int g = nt*16 + col;

        // B fragments: one b64 load per chunk (pre-paired layout)
        const v2f b0 = B0[g];
        const v2f b1 = B1[g];
        const v2f b2 = B2[g];

        v8f acc = {};
        acc = __builtin_amdgcn_wmma_f32_16x16x4_f32(false, a0, false, b0,
                                                    (short)0, acc, false, false);
        acc = __builtin_amdgcn_wmma_f32_16x16x4_f32(false, a1, false, b1,
                                                    (short)0, acc, false, false);
        acc = __builtin_amdgcn_wmma_f32_16x16x4_f32(false, a2, false, b2,
                                                    (short)0, acc, false, false);

        // Features: one b128 load
        const v4f fv = feat4[g];

        // D layout: this lane holds gaussian column `g`, pixel rows
        // half*8 + r (r = VGPR index). w = exp(-sigma); rgb += w * feat.
#pragma unroll
        for (int r = 0; r < 8; ++r) {
            const float w = __expf(-acc[r]);
            rgb[r][0] += w * fv.x;
            rgb[r][1] += w * fv.y;
            rgb[r][2] += w * fv.z;
        }
    }

    // ---- reduce the 16 gaussian columns per pixel via per-wave LDS slice ----
    float* my = lds[wv];
#pragma unroll
    for (int r = 0; r < 8; ++r) {
        const int pr = half*8 + r;
        my[(pr*3+0)*16 + col] = rgb[r][0];
        my[(pr*3+1)*16 + col] = rgb[r][1];
        my[(pr*3+2)*16 + col] = rgb[r][2];
    }
    // Same-wave LDS RAW: DS ops are in-order per wave; just drain DScnt.
    asm volatile("s_wait_dscnt 0" ::: "memory");

    // 48 (pixRow, ch) outputs per wave; lanes cover q = lane and lane+32.
#pragma unroll
    for (int it = 0; it < 2; ++it) {
        const int q = lane + it*32;
        if (q < 48) {
            const int pr = q / 3, ch = q % 3;
            float s = 0.f;
#pragma unroll
            for (int c = 0; c < 16; ++c) s += my[(pr*3+ch)*16 + c];
            s = fminf(fmaxf(s, 0.0f), 1.0f);
            const int pp = pbase + pr;
            const int xx = pp & (Wd-1);
            const int yy = (pp >> 7) & (Hd-1);
            const int tt = pp >> 14;
            // (1,3,H,W,T) layout
            out[((ch*Hd + yy)*Wd + xx)*Td + tt] = s;
        }
    }
}

extern "C" void kernel_launch(void* const* d_in, const int* in_sizes, int n_in,
                              void* d_out, int out_size, void* d_ws, size_t ws_size,
                              hipStream_t stream) {
    const float* xyz      = (const float*)d_in[0];
    const float* chol     = (const float*)d_in[1];
    const float* features = (const float*)d_in[2];
    const float* opacity  = (const float*)d_in[3];
    // d_in[4] = background, unused by the sum-rasterizer

    v2f* Gp    = (v2f*)d_ws;           // 6*1024 float2 = 48 KB
    v4f* feat4 = (v4f*)((char*)d_ws + 6*NG*sizeof(v2f));  // 1024 float4 = 16 KB
    float* out = (float*)d_out;        // 3*128*128*8 floats

    prep_kernel<<<(NG + 255)/256, 256, 0, stream>>>(xyz, chol, features, opacity,
                                                    Gp, feat4);

    // P/16 = 8192 waves, 8 waves (256 threads) per block -> 1024 blocks
    render_kernel<<<(Td*Hd*Wd)/(16*8), 256, 0, stream>>>(Gp, feat4, out);
}